// RBMConv_65042984730852
// MI455X (gfx1250) — compile-verified
//
#include <hip/hip_runtime.h>
#include <math.h>

// CRBM CD-1 sweep for MI455X (gfx1250, wave32).
// Forward conv + gradient correlations run on V_WMMA_F32_16X16X4_F32 (f32 in/out,
// reference precision). Inner loops are branch-free (incremental addressing,
// mask-multiply tail) so EXEC stays all-1s around WMMA. Gradient GEMM uses
// split-K (32 segments) with a deterministic fixed-order reduction.

typedef __attribute__((ext_vector_type(2))) float v2f;
typedef __attribute__((ext_vector_type(8))) float v8f;

#define NVD 128
#define NHD 112
#define KF  32
#define FHW 17
#define RF  289       // FHW*FHW taps
#define NPIX 16384    // NVD*NVD
#define NHID 12544    // NHD*NHD
#define LRATE 0.01f

#define NSEG 32       // split-K segments for gradient GEMM
#define SEGR (NHID / NSEG)   // 392 positions per segment (392 % 4 == 0)
#define MPAD 304      // 19 M-tiles * 16 (padded 289)

// workspace layout (float offsets)
#define WS_W 0                       // 9248 floats
#define WS_A 9248                    // 1 float
#define WS_B 9252                    // 32 floats
#define WS_H 9296                    // 401408 floats (16B aligned)
#define WS_V (WS_H + 401408)         // 16384 floats
#define WS_P (WS_V + NPIX)           // 32*2*304*32 = 622592 floats

__device__ __forceinline__ float sigm(float x) { return 1.0f / (1.0f + expf(-x)); }

__global__ void k_init(const float* __restrict__ Wi, const float* __restrict__ ai,
                       const float* __restrict__ bi, float* __restrict__ ws) {
    int t = blockIdx.x * blockDim.x + threadIdx.x;
    if (t < RF * KF) ws[WS_W + t] = Wi[t];
    if (t == 0)      ws[WS_A]     = ai[0];
    if (t < KF)      ws[WS_B + t] = bi[t];
}

// hprob[m,k] = sigmoid(b[k] + sum_r v0patch[m][r] * Wflip[r][k]); M=12544,N=32,K=289
__global__ void k_fwd(const float* __restrict__ inputs, float* __restrict__ ws, int sample) {
    const float* v0 = inputs + sample * NPIX;
    const float* Wc = ws + WS_W;
    const float* bb = ws + WS_B;
    float* hprob = ws + WS_H;

    int wid  = blockIdx.x * (blockDim.x >> 5) + (threadIdx.x >> 5); // 0..783
    int lane = threadIdx.x & 31;
    int half = lane >> 4;
    int lm   = lane & 15;

    int m = wid * 16 + lm;              // A-row (hidden position), same for both halves
    int y = m / NHD, x = m % NHD;
    int base = y * NVD + x;

    // incremental tap addressing: r0 = kk + 2*half, r1 = r0 + 1 (lane-half K slots)
    int j0 = 2 * half, j1 = j0 + 1;     // tap column within filter row (i starts 0)
    int addr0 = base + j0, addr1 = base + j1;
    int w0 = (RF - 1 - 2 * half) * KF;  // Wflip index = (288 - r)*32 (linear -128/step)
    int w1 = w0 - KF;
    const float* Wl = Wc + lm;

    v8f c0 = {}; v8f c1 = {};
    #pragma unroll 2
    for (int it = 0; it < 72; ++it) {   // covers r in [0,288), branch-free
        v2f a, b0, b1;
        a.x  = v0[addr0];      a.y  = v0[addr1];
        b0.x = Wl[w0];         b0.y = Wl[w1];
        b1.x = Wl[w0 + 16];    b1.y = Wl[w1 + 16];
        c0 = __builtin_amdgcn_wmma_f32_16x16x4_f32(false, a, false, b0, (short)0, c0, false, false);
        c1 = __builtin_amdgcn_wmma_f32_16x16x4_f32(false, a, false, b1, (short)0, c1, false, false);
        addr0 += 4; j0 += 4; if (j0 >= FHW) { j0 -= FHW; addr0 += NVD - FHW; }
        addr1 += 4; j1 += 4; if (j1 >= FHW) { j1 -= FHW; addr1 += NVD - FHW; }
        w0 -= 4 * KF; w1 -= 4 * KF;
    }
    { // tail: r = 288 only (K-slot 0, lane-half 0); zero via mask multiply, no EXEC churn
        float mk = (half == 0) ? 1.0f : 0.0f;
        v2f a, b0, b1;
        a.x  = v0[base + 16 * NVD + 16] * mk;  a.y = 0.0f;
        b0.x = Wl[0];       b0.y = 0.0f;       // Wflip[288] = W[0]
        b1.x = Wl[16];      b1.y = 0.0f;
        c0 = __builtin_amdgcn_wmma_f32_16x16x4_f32(false, a, false, b0, (short)0, c0, false, false);
        c1 = __builtin_amdgcn_wmma_f32_16x16x4_f32(false, a, false, b1, (short)0, c1, false, false);
    }
    // C/D layout: VGPR j holds rows j (lanes 0-15) and j+8 (lanes 16-31), col = lane&15
    #pragma unroll
    for (int jv = 0; jv < 8; ++jv) {
        int mm = wid * 16 + jv + 8 * half;
        int n0 = lm, n1 = lm + 16;
        hprob[mm * KF + n0] = sigm(bb[n0] + c0[jv]);
        hprob[mm * KF + n1] = sigm(bb[n1] + c1[jv]);
    }
}

// v[p] = sigmoid(a + sum_{s,t,k} hprob[p-s, q-t, k] * W[s,t,k])  (conv transpose, N=1 -> VALU)
__global__ void k_recon(float* __restrict__ ws) {
    int p = blockIdx.x * blockDim.x + threadIdx.x;
    if (p >= NPIX) return;
    const float* Wc = ws + WS_W;
    const float* h  = ws + WS_H;
    float acc = ws[WS_A];
    int py = p / NVD, px = p % NVD;
    for (int s = 0; s < FHW; ++s) {
        int hy = py - s; if ((unsigned)hy >= (unsigned)NHD) continue;
        for (int t = 0; t < FHW; ++t) {
            int hx = px - t; if ((unsigned)hx >= (unsigned)NHD) continue;
            const float4* hv = (const float4*)(h + (hy * NHD + hx) * KF);
            const float4* wv = (const float4*)(Wc + (s * FHW + t) * KF);
            #pragma unroll
            for (int q = 0; q < 8; ++q) {
                float4 hq = hv[q], wq = wv[q];
                acc = fmaf(hq.x, wq.x, acc); acc = fmaf(hq.y, wq.y, acc);
                acc = fmaf(hq.z, wq.z, acc); acc = fmaf(hq.w, wq.w, acc);
            }
        }
    }
    ws[WS_V + p] = sigm(acc);
}

// Split-K gradient GEMM: partial g0/gt over one 392-position segment.
// grid (19, 32): blockIdx.x = mtile, blockIdx.y = seg; block 64 = 2 waves (ntile 0/1).
__global__ void k_grad(const float* __restrict__ inputs, const float* __restrict__ u1,
                       const float* __restrict__ u2, float* __restrict__ ws, int sample) {
    const float* v0 = inputs + sample * NPIX;
    const float* vr = ws + WS_V;
    const float* hprob = ws + WS_H;
    const float* U1 = u1 + (size_t)sample * NHID * KF;
    const float* U2 = u2 + (size_t)sample * NHID * KF;

    int mtile = blockIdx.x;             // 0..18
    int seg   = blockIdx.y;             // 0..31
    int ntile = threadIdx.x >> 5;       // 0..1
    int lane  = threadIdx.x & 31;
    int half  = lane >> 4;
    int lm    = lane & 15;

    int fm  = mtile * 16 + lm;          // A-row = filter tap (i,j); clamp OOB (rows unused)
    int fmc = fm < RF ? fm : RF - 1;
    int fi = fmc / FHW, fj = fmc % FHW;
    int n = ntile * 16 + lm;            // filter channel column

    int r0 = seg * SEGR + 2 * half;     // inner position index for K-slot 0 (slot1 = r0+1)
    int y0 = r0 / NHD, xc = r0 % NHD;   // xc even -> r0+1 never wraps a row
    int addrA = (y0 + fi) * NVD + xc + fj;
    int hIdx = r0 * KF + n;             // h/u index, linear +128/step

    v8f cg0 = {}; v8f cgt = {};
    #pragma unroll 2
    for (int it = 0; it < SEGR / 4; ++it) {   // 98 branch-free K-steps
        v2f a0, a1, b0, b1;
        a0.x = v0[addrA];      a0.y = v0[addrA + 1];
        a1.x = vr[addrA];      a1.y = vr[addrA + 1];
        float hp0 = hprob[hIdx], hp1 = hprob[hIdx + KF];
        b0.x = (hp0 > U2[hIdx])      ? 1.f : 0.f;   // h_init (vs u2)
        b0.y = (hp1 > U2[hIdx + KF]) ? 1.f : 0.f;
        b1.x = (hp0 > U1[hIdx])      ? 1.f : 0.f;   // h_bin (vs u1)
        b1.y = (hp1 > U1[hIdx + KF]) ? 1.f : 0.f;
        cg0 = __builtin_amdgcn_wmma_f32_16x16x4_f32(false, a0, false, b0, (short)0, cg0, false, false);
        cgt = __builtin_amdgcn_wmma_f32_16x16x4_f32(false, a1, false, b1, (short)0, cgt, false, false);
        addrA += 4; xc += 4; if (xc >= NHD) { xc -= NHD; addrA += NVD - NHD; }
        hIdx += 4 * KF;
    }
    float* P0 = ws + WS_P + (size_t)(seg * 2 + 0) * MPAD * KF;
    float* P1 = ws + WS_P + (size_t)(seg * 2 + 1) * MPAD * KF;
    #pragma unroll
    for (int jv = 0; jv < 8; ++jv) {
        int mm = mtile * 16 + jv + 8 * half;   // < MPAD, unguarded (padded rows)
        P0[mm * KF + n] = cg0[jv];
        P1[mm * KF + n] = cgt[jv];
    }
}

// W += LR * sum_seg (g0_partial - gt_partial), fixed summation order (deterministic)
__global__ void k_wreduce(float* __restrict__ ws) {
    int t = blockIdx.x * blockDim.x + threadIdx.x;
    if (t >= RF * KF) return;
    float acc = 0.f;
    for (int s = 0; s < NSEG; ++s) {
        const float* base = ws + WS_P + (size_t)s * 2 * MPAD * KF;
        acc += base[t] - base[MPAD * KF + t];
    }
    ws[WS_W + t] += LRATE * acc;
}

// a += LR*(sum(v0)-sum(v));  b[k] += LR * sum(h_init - h_bin)  (integer-exact)
__global__ void k_update(const float* __restrict__ inputs, const float* __restrict__ u1,
                         const float* __restrict__ u2, float* __restrict__ ws, int sample) {
    __shared__ float smf[1024];
    int t = threadIdx.x;
    const float* v0 = inputs + sample * NPIX;
    const float* vr = ws + WS_V;

    float s = 0.f;
    for (int i = t; i < NPIX; i += 1024) s += v0[i];
    smf[t] = s; __syncthreads();
    for (int off = 512; off > 0; off >>= 1) { if (t < off) smf[t] += smf[t + off]; __syncthreads(); }
    float S0 = smf[0]; __syncthreads();

    s = 0.f;
    for (int i = t; i < NPIX; i += 1024) s += vr[i];
    smf[t] = s; __syncthreads();
    for (int off = 512; off > 0; off >>= 1) { if (t < off) smf[t] += smf[t + off]; __syncthreads(); }
    if (t == 0) ws[WS_A] += LRATE * (S0 - smf[0]);
    __syncthreads();

    int* smi = (int*)smf;
    const float* hprob = ws + WS_H;
    const float* U1 = u1 + (size_t)sample * NHID * KF;
    const float* U2 = u2 + (size_t)sample * NHID * KF;
    int k = t & 31, g = t >> 5;
    int isum = 0;
    for (int p = g; p < NHID; p += 32) {
        int idx = p * KF + k;
        float hp = hprob[idx];
        isum += (hp > U2[idx] ? 1 : 0) - (hp > U1[idx] ? 1 : 0);
    }
    smi[t] = isum; __syncthreads();
    for (int off = 512; off >= 32; off >>= 1) { if (t < off) smi[t] += smi[t + off]; __syncthreads(); }
    if (t < 32) ws[WS_B + t] += LRATE * (float)smi[t];
}

__global__ void k_copy(const float* __restrict__ ws, float* __restrict__ out, int n) {
    int t = blockIdx.x * blockDim.x + threadIdx.x;
    if (t < n) out[t] = ws[WS_H + t];
}

extern "C" void kernel_launch(void* const* d_in, const int* in_sizes, int n_in,
                              void* d_out, int out_size, void* d_ws, size_t ws_size,
                              hipStream_t stream) {
    const float* inputs = (const float*)d_in[0];   // (32,128,128,1)
    const float* u1     = (const float*)d_in[1];   // (32,1,112,112,32)
    const float* u2     = (const float*)d_in[2];   // (32,1,112,112,32)
    const float* Wi     = (const float*)d_in[3];   // (17,17,1,32)
    const float* ai     = (const float*)d_in[4];   // scalar
    const float* bi     = (const float*)d_in[5];   // (32,)
    float* ws = (float*)d_ws;

    k_init<<<37, 256, 0, stream>>>(Wi, ai, bi, ws);
    for (int smp = 0; smp < 32; ++smp) {
        k_fwd  <<<196, 128, 0, stream>>>(inputs, ws, smp);          // 784 waves, WMMA f32
        k_recon<<<64, 256, 0, stream>>>(ws);                        // VALU conv-transpose
        k_grad <<<dim3(19, 32), 64, 0, stream>>>(inputs, u1, u2, ws, smp); // 1216 waves, split-K WMMA
        k_wreduce<<<37, 256, 0, stream>>>(ws);                      // deterministic seg-sum + W update
        k_update<<<1, 1024, 0, stream>>>(inputs, u1, u2, ws, smp);
    }
    k_copy<<<(401408 + 1023) / 1024, 1024, 0, stream>>>(ws, (float*)d_out, 401408);
}